// IGMDALayer_30064771072069
// MI455X (gfx1250) — compile-verified
//
#include <hip/hip_runtime.h>

typedef __attribute__((ext_vector_type(2))) float v2f;
typedef __attribute__((ext_vector_type(8))) float v8f;

#define N1 8192
#define N2 8192
#define FM 256
#define FOUT 128

// d_ws layout (floats): [0..255]=v_m  [256..511]=v_d  [512..8703]=s1  [8704..16895]=s2

// ---------------- Kernel 1: fold v = W @ a_slice (tiny) ----------------
__global__ void fold_w_a_kernel(const float* __restrict__ w_m,
                                const float* __restrict__ w_d,
                                const float* __restrict__ a,
                                float* __restrict__ ws) {
    int t = threadIdx.x;               // 512 threads, 1 block
    const float* wrow;
    const float* avec;
    int dst;
    if (t < 256) { wrow = w_m + t * FOUT;          avec = a;        dst = t; }
    else         { wrow = w_d + (t - 256) * FOUT;  avec = a + FOUT; dst = t; }
    float acc = 0.f;
    #pragma unroll 8
    for (int j = 0; j < FOUT; ++j) acc += wrow[j] * avec[j];
    ws[dst] = acc;
}

// ---------------- Kernel 2: s1/s2 GEMV, one wave32 per row ----------------
__global__ void row_scores_kernel(const float* __restrict__ m,  const float* __restrict__ m1,
                                  const float* __restrict__ m2, const float* __restrict__ d,
                                  const float* __restrict__ d1, const float* __restrict__ d2,
                                  const float* __restrict__ a_m, const float* __restrict__ a_d,
                                  float* __restrict__ ws) {
    const int lane = threadIdx.x & 31;
    const int wave = threadIdx.x >> 5;            // 8 waves / block
    const bool isD = (blockIdx.x >= 1024);        // blocks 0..1023 -> s1, 1024..2047 -> s2
    const int  blk = isD ? (blockIdx.x - 1024) : blockIdx.x;
    const int  row = blk * 8 + wave;              // 1024*8 = 8192 rows each side

    const float* A   = isD ? d  : m;
    const float* B   = isD ? d1 : m1;
    const float* C   = isD ? d2 : m2;
    const float* mix = isD ? a_d : a_m;
    const float* v   = ws + (isD ? 256 : 0);

    const float c0 = mix[0], c1 = mix[1], c2 = mix[2];
    const size_t base = (size_t)row * FM;

    float acc = 0.f;
    #pragma unroll
    for (int i = 0; i < FM / 32; ++i) {           // lane-stride-32: coalesced
        int col = lane + i * 32;
        float h = c0 * A[base + col] + c1 * B[base + col] + c2 * C[base + col];
        acc += h * v[col];
    }
    #pragma unroll
    for (int off = 16; off >= 1; off >>= 1)       // wave32 tree reduce
        acc += __shfl_xor(acc, off, 32);
    if (lane == 0) ws[512 + (isD ? 8192 : 0) + row] = acc;
}

// ------- Kernel 3: 16x16 tile = WMMA rank-2 outer sum, then mish, store -------
// A(16x4): col0 = s1_tile, col1 = ones, cols 2,3 = 0  (lanes 16..31 hold K=2,3 -> zero)
// B(4x16): row0 = ones, row1 = s2_tile, rows 2,3 = 0  (lanes 16..31 hold K=2,3 -> zero)
// D[mrow][ncol] = s1[mrow] + s2[ncol]
__global__ __launch_bounds__(256)
void outer_mish_kernel(const float* __restrict__ ws, float* __restrict__ out) {
    const int lane   = threadIdx.x & 31;
    const int wave   = threadIdx.x >> 5;
    const int tileId = blockIdx.x * 8 + wave;     // 32768 blocks * 8 = 512*512 tiles
    const int tm = tileId >> 9;                   // tile row   (N1/16 = 512)
    const int tn = tileId & 511;                  // tile col   (N2/16 = 512)

    const float* s1 = ws + 512;
    const float* s2 = ws + 8704;

    const int l16  = lane & 15;
    const float s1v = s1[tm * 16 + l16];          // broadcast L2 hits (tiny arrays)
    const float s2v = s2[tn * 16 + l16];
    const bool  lo  = (lane < 16);                // cndmask select, no divergence

    v2f A, B;
    A.x = lo ? s1v : 0.f;   A.y = lo ? 1.f : 0.f; // A: K=0 / K=1 (lanes>=16: K=2,3 -> 0)
    B.x = lo ? 1.f  : 0.f;  B.y = lo ? s2v : 0.f; // B: K=0 / K=1 (lanes>=16: K=2,3 -> 0)
    v8f C = {};

    // v_wmma_f32_16x16x4_f32: D = A x B + 0  -> full 16x16 outer-sum tile
    v8f D = __builtin_amdgcn_wmma_f32_16x16x4_f32(
        /*neg_a=*/false, A, /*neg_b=*/false, B,
        /*c_mod=*/(short)0, C, /*reuse_a=*/false, /*reuse_b=*/false);

    // mish(x) = x*tanh(softplus(x)) = x*(t^2-1)/(t^2+1), t = 1+e^x  (one exp, one rcp)
    const float LOG2E = 1.4426950408889634f;
    const int colBase = tn * 16 + l16;
    const int rowBase = tm * 16 + (lo ? 0 : 8);   // C/D layout: VGPR r -> rows r / r+8
    #pragma unroll
    for (int r = 0; r < 8; ++r) {
        float x  = D[r];
        float ex = __builtin_amdgcn_exp2f(fminf(x, 40.f) * LOG2E); // clamp avoids inf/inf
        float t  = 1.f + ex;
        float n2 = t * t;
        float mv = x * (n2 - 1.f) * __builtin_amdgcn_rcpf(n2 + 1.f);
        out[(size_t)(rowBase + r) * N2 + colBase] = mv;  // 64B coalesced per half-wave
    }
}

extern "C" void kernel_launch(void* const* d_in, const int* in_sizes, int n_in,
                              void* d_out, int out_size, void* d_ws, size_t ws_size,
                              hipStream_t stream) {
    const float* m   = (const float*)d_in[0];
    const float* m1  = (const float*)d_in[1];
    const float* m2  = (const float*)d_in[2];
    const float* d   = (const float*)d_in[3];
    const float* d1  = (const float*)d_in[4];
    const float* d2  = (const float*)d_in[5];
    // d_in[6] = adj (unused by forward)
    const float* w_m = (const float*)d_in[7];
    const float* w_d = (const float*)d_in[8];
    const float* a_m = (const float*)d_in[9];
    const float* a_d = (const float*)d_in[10];
    const float* a   = (const float*)d_in[11];

    float* ws  = (float*)d_ws;
    float* out = (float*)d_out;

    fold_w_a_kernel  <<<1,     512, 0, stream>>>(w_m, w_d, a, ws);
    row_scores_kernel<<<2048,  256, 0, stream>>>(m, m1, m2, d, d1, d2, a_m, a_d, ws);
    outer_mish_kernel<<<32768, 256, 0, stream>>>(ws, out);
}